// PolyGCLayer_21182778704682
// MI455X (gfx1250) — compile-verified
//
#include <hip/hip_runtime.h>

typedef __attribute__((ext_vector_type(2))) float v2f;
typedef __attribute__((ext_vector_type(8))) float v8f;

#define N_NODES 4096
#define CIN     32
#define BATCH   32
#define CB      1024      // CIN * BATCH
#define NNZ_TOT 65536
#define KDEG    5
#define COUT    64
#define MROWS   131072    // BATCH * N_NODES
#define KDIM    160       // CIN * KDEG

// ---------------------------------------------------------------------------
// 1. Pack x [B,N,C] -> x0 [N, C*B] and also write A column block for k=0
//    (A is stored column-major: A[col * MROWS + b*N + n], col = c*KDEG + k)
// ---------------------------------------------------------------------------
__global__ __launch_bounds__(256) void pack_x0_kernel(
    const float* __restrict__ x, float* __restrict__ xb0, float* __restrict__ A) {
  int tid = blockIdx.x * 256 + threadIdx.x;          // 4,194,304 threads, n fastest
  int n = tid & (N_NODES - 1);
  int b = (tid >> 12) & 31;
  int c = tid >> 17;
  float v = x[((size_t)(b * N_NODES + n)) * CIN + c];
  xb0[(size_t)n * CB + c * BATCH + b] = v;
  A[((size_t)(c * KDEG + 0)) * MROWS + (size_t)b * N_NODES + n] = v;
}

// ---------------------------------------------------------------------------
// 2. Deterministic CSR build (stable chunked counting sort, 256 chunks x 256)
// ---------------------------------------------------------------------------
__global__ __launch_bounds__(256) void csr_chunk_count(
    const int* __restrict__ erow, int* __restrict__ cnt2) {
  __shared__ int hist[N_NODES];
  int t = threadIdx.x, chunk = blockIdx.x;
  for (int i = t; i < N_NODES; i += 256) hist[i] = 0;
  __syncthreads();
  atomicAdd(&hist[erow[chunk * 256 + t]], 1);        // LDS atomics: count only
  __syncthreads();
  for (int i = t; i < N_NODES; i += 256)
    cnt2[(size_t)chunk * N_NODES + i] = hist[i];
}

__global__ __launch_bounds__(256) void csr_scan_rows(
    int* __restrict__ cnt2, int* __restrict__ rowptr) {
  int row = blockIdx.x * 256 + threadIdx.x;          // 4096 threads
  int total = 0;
  for (int ch = 0; ch < 256; ++ch) {
    size_t idx = (size_t)ch * N_NODES + row;
    int v = cnt2[idx];
    cnt2[idx] = total;                               // exclusive prefix over chunks
    total += v;
  }
  rowptr[row] = total;                               // row totals (scanned next)
}

__global__ void csr_scan_global(int* __restrict__ rowptr) {
  int run = 0;
  for (int i = 0; i < N_NODES; ++i) { int v = rowptr[i]; rowptr[i] = run; run += v; }
  rowptr[N_NODES] = run;
}

__global__ __launch_bounds__(256) void csr_scatter(
    const int* __restrict__ erow, const int* __restrict__ rowptr,
    const int* __restrict__ cnt2, int* __restrict__ perm) {
  __shared__ int rows[256];
  int t = threadIdx.x, chunk = blockIdx.x;
  int e = chunk * 256 + t;
  int r = erow[e];
  rows[t] = r;
  __syncthreads();
  int rank = 0;                                      // stable rank within chunk
  for (int i = 0; i < t; ++i) rank += (rows[i] == r) ? 1 : 0;
  perm[rowptr[r] + cnt2[(size_t)chunk * N_NODES + r] + rank] = e;
}

// ---------------------------------------------------------------------------
// 3. CSR SPMM + Chebyshev combine: dst = scale * (L @ src) - sub
//    One block per row; thread j owns a float4 slice of the 1024-wide row.
// ---------------------------------------------------------------------------
__global__ __launch_bounds__(256) void spmm_cheb(
    float* __restrict__ dst, const float* __restrict__ src,
    const float* __restrict__ sub, float scale,
    const int* __restrict__ rowptr, const int* __restrict__ perm,
    const float* __restrict__ eval, const int* __restrict__ ecol) {
  int row = blockIdx.x;
  int off = threadIdx.x * 4;
  int s = rowptr[row], e = rowptr[row + 1];
  float4 acc = make_float4(0.f, 0.f, 0.f, 0.f);
  for (int p = s; p < e; ++p) {
    int ei  = perm[p];
    float v = eval[ei];
    int col = ecol[ei];
    if (p + 1 < e) {                                 // prefetch next gathered row
      int cn = ecol[perm[p + 1]];
      __builtin_prefetch(src + (size_t)cn * CB + off, 0, 0);
    }
    const float4 g = *(const float4*)(src + (size_t)col * CB + off);
    acc.x += v * g.x; acc.y += v * g.y; acc.z += v * g.z; acc.w += v * g.w;
  }
  float4 sb = make_float4(0.f, 0.f, 0.f, 0.f);
  if (sub) sb = *(const float4*)(sub + (size_t)row * CB + off);
  float4 r;
  r.x = scale * acc.x - sb.x; r.y = scale * acc.y - sb.y;
  r.z = scale * acc.z - sb.z; r.w = scale * acc.w - sb.w;
  *(float4*)(dst + (size_t)row * CB + off) = r;
}

// ---------------------------------------------------------------------------
// 4. Pack x_k [N, C*B] into column-major A at columns c*KDEG + k
// ---------------------------------------------------------------------------
__global__ __launch_bounds__(256) void pack_Acol(
    const float* __restrict__ xb, float* __restrict__ A, int k) {
  int tid = blockIdx.x * 256 + threadIdx.x;          // n fastest -> coalesced A write
  int n  = tid & (N_NODES - 1);
  int cb = tid >> 12;
  int c = cb >> 5, b = cb & 31;
  A[((size_t)(c * KDEG + k)) * MROWS + (size_t)b * N_NODES + n] =
      xb[(size_t)n * CB + cb];
}

// ---------------------------------------------------------------------------
// 5. GEMM [131072 x 160] @ [160 x 64] via V_WMMA_F32_16X16X4_F32,
//    fused bias + ReLU + maxpool(4 nodes). One wave = 16-row strip x 64 cols.
// ---------------------------------------------------------------------------
__global__ __launch_bounds__(256) void gemm_wmma_epilogue(
    const float* __restrict__ A, const float* __restrict__ W,
    const float* __restrict__ bias, float* __restrict__ out) {
  int wave = threadIdx.x >> 5;
  int lane = threadIdx.x & 31;
  int half = lane >> 4;          // 0: lanes 0-15, 1: lanes 16-31
  int lr   = lane & 15;
  int m0   = (blockIdx.x * 8 + wave) << 4;

  v8f acc0 = {}, acc1 = {}, acc2 = {}, acc3 = {};

#pragma unroll
  for (int kt = 0; kt < KDIM / 4; ++kt) {
    int ca = kt * 4 + half * 2;
    // A fragment 16x4 f32: lane half selects K pair, lr selects row M.
    v2f a;
    a.x = A[(size_t)(ca + 0) * MROWS + m0 + lr];
    a.y = A[(size_t)(ca + 1) * MROWS + m0 + lr];
    // B fragments 4x16 f32 (row-striped across lanes, mirrored K pairing).
    const float* w0 = W + (ca + 0) * COUT + lr;
    const float* w1 = W + (ca + 1) * COUT + lr;
    v2f b0, b1, b2, b3;
    b0.x = w0[0];  b0.y = w1[0];
    b1.x = w0[16]; b1.y = w1[16];
    b2.x = w0[32]; b2.y = w1[32];
    b3.x = w0[48]; b3.y = w1[48];
    acc0 = __builtin_amdgcn_wmma_f32_16x16x4_f32(false, a, false, b0, (short)0, acc0, false, false);
    acc1 = __builtin_amdgcn_wmma_f32_16x16x4_f32(false, a, false, b1, (short)0, acc1, false, false);
    acc2 = __builtin_amdgcn_wmma_f32_16x16x4_f32(false, a, false, b2, (short)0, acc2, false, false);
    acc3 = __builtin_amdgcn_wmma_f32_16x16x4_f32(false, a, false, b3, (short)0, acc3, false, false);
  }

  // Epilogue: bias + ReLU + maxpool over 4 consecutive node rows.
  // C layout: VGPR r holds M=r (lanes 0-15) / M=r+8 (lanes 16-31), N=lr.
  int n0 = m0 & (N_NODES - 1);
  int bb = m0 >> 12;
  int orow = bb * (N_NODES / 4) + (n0 >> 2);
  int g0 = half * 2, g1 = half * 2 + 1;

#define EPILOG(ACC, T)                                                        \
  {                                                                           \
    int col = (T) * 16 + lr;                                                  \
    float bv = bias[col];                                                     \
    float e0 = fmaxf(ACC[0] + bv, 0.f), e1 = fmaxf(ACC[1] + bv, 0.f);         \
    float e2 = fmaxf(ACC[2] + bv, 0.f), e3 = fmaxf(ACC[3] + bv, 0.f);         \
    float e4 = fmaxf(ACC[4] + bv, 0.f), e5 = fmaxf(ACC[5] + bv, 0.f);         \
    float e6 = fmaxf(ACC[6] + bv, 0.f), e7 = fmaxf(ACC[7] + bv, 0.f);         \
    float p0 = fmaxf(fmaxf(e0, e1), fmaxf(e2, e3));                           \
    float p1 = fmaxf(fmaxf(e4, e5), fmaxf(e6, e7));                           \
    out[(size_t)(orow + g0) * COUT + col] = p0;                               \
    out[(size_t)(orow + g1) * COUT + col] = p1;                               \
  }
  EPILOG(acc0, 0)
  EPILOG(acc1, 1)
  EPILOG(acc2, 2)
  EPILOG(acc3, 3)
#undef EPILOG
}

// ---------------------------------------------------------------------------
// Launch
// ---------------------------------------------------------------------------
extern "C" void kernel_launch(void* const* d_in, const int* in_sizes, int n_in,
                              void* d_out, int out_size, void* d_ws, size_t ws_size,
                              hipStream_t stream) {
  const float* x        = (const float*)d_in[0];
  const float* edge_val = (const float*)d_in[1];
  const float* weight   = (const float*)d_in[2];
  const float* bias     = (const float*)d_in[3];
  const int*   edge_row = (const int*)d_in[4];
  const int*   edge_col = (const int*)d_in[5];
  float* out = (float*)d_out;

  // Workspace carve-up (all offsets 256B-aligned).
  char* ws = (char*)d_ws;
  size_t off = 0;
  int* rowptr = (int*)(ws + off);           off += 32768;                 // 4097 ints
  int* cnt2   = (int*)(ws + off);           off += (size_t)256 * N_NODES * 4; // 4 MB
  int* perm   = (int*)(ws + off);           off += (size_t)NNZ_TOT * 4;   // 256 KB
  float* xb   = (float*)(ws + off);         off += (size_t)3 * N_NODES * CB * 4; // 48 MB
  float* A    = (float*)(ws + off);         off += (size_t)KDIM * MROWS * 4;     // 80 MB
  if (ws_size < off) return;  // deterministic no-op if scratch too small

  float* xbuf[3] = {xb, xb + (size_t)N_NODES * CB, xb + (size_t)2 * N_NODES * CB};

  // 1. pack x -> x0 (and A columns for k=0)
  pack_x0_kernel<<<(N_NODES * CB) / 256, 256, 0, stream>>>(x, xbuf[0], A);

  // 2. deterministic CSR build
  csr_chunk_count<<<256, 256, 0, stream>>>(edge_row, cnt2);
  csr_scan_rows<<<N_NODES / 256, 256, 0, stream>>>(cnt2, rowptr);
  csr_scan_global<<<1, 1, 0, stream>>>(rowptr);
  csr_scatter<<<256, 256, 0, stream>>>(edge_row, rowptr, cnt2, perm);

  // 3. Chebyshev recursion: x1 = L@x0 ; x_k = 2*(L@x_{k-1}) - x_{k-2}
  for (int k = 1; k < KDEG; ++k) {
    float* dst = xbuf[k % 3];
    const float* src = xbuf[(k - 1) % 3];
    const float* sub = (k >= 2) ? xbuf[(k - 2) % 3] : nullptr;
    float scale = (k >= 2) ? 2.0f : 1.0f;
    spmm_cheb<<<N_NODES, 256, 0, stream>>>(dst, src, sub, scale,
                                           rowptr, perm, edge_val, edge_col);
    pack_Acol<<<(N_NODES * CB) / 256, 256, 0, stream>>>(dst, A, k);
  }

  // 4. fused WMMA GEMM + bias + ReLU + maxpool(4)
  gemm_wmma_epilogue<<<MROWS / 16 / 8, 256, 0, stream>>>(A, weight, bias, out);
}